// QuantizationLayer_71820443124073
// MI455X (gfx1250) — compile-verified
//
#include <hip/hip_runtime.h>
#include <hip/hip_bf16.h>

typedef float v2f __attribute__((ext_vector_type(2)));
typedef float v4f __attribute__((ext_vector_type(4)));
typedef float v8f __attribute__((ext_vector_type(8)));

#if defined(__has_builtin)
#if __has_builtin(__builtin_amdgcn_sched_barrier)
#define SCHED_FENCE() __builtin_amdgcn_sched_barrier(0)
#endif
#endif
#ifndef SCHED_FENCE
#define SCHED_FENCE()
#endif

#define NROWS 131072
#define DIMSZ 128
#define KCB   4096

// ---------------- ws layout (floats) ----------------
#define WS_CNT   4096
#define WS_ESUM  8192
#define WS_ACC   532480
#define WS_ZERO_LEN (WS_ACC + 3 - WS_CNT)

// ---------------- d_out layout (floats) ----------------
#define O_Q      0
#define O_LOSS   16777216
#define O_SMALL  16777217
#define O_IND    16777218
#define O_ENEW   16908290
#define O_CSNEW  17432578
#define O_EANEW  17436674

__global__ void k_init(float* __restrict__ ws) {
    int i = blockIdx.x * 256 + threadIdx.x;
    if (i < WS_ZERO_LEN) ws[WS_CNT + i] = 0.0f;
}

__global__ void k_norm(const float* __restrict__ embed, float* __restrict__ half_norm) {
    int k = blockIdx.x * 256 + threadIdx.x;
    float s = 0.0f;
    #pragma unroll 8
    for (int d = 0; d < DIMSZ; ++d) { float e = embed[d * KCB + k]; s += e * e; }
    half_norm[k] = 0.5f * s;
}

// Fused x@embed GEMM (f32 WMMA) + argmax(score).
// - LDS embed tile double-buffered against global prefetch (float4).
// - B fragments software-pipelined in batches of 8 with sched fences, so each
//   WMMA batch consumes fragments loaded one full batch earlier (dscnt<=8).
// - Two independent accumulator chains per wave.
__global__ __launch_bounds__(256, 1)
void k_assign(const float* __restrict__ x, const float* __restrict__ embed,
              const float* __restrict__ half_norm,
              float* __restrict__ ind_out, float* __restrict__ cnt) {
    __shared__ float tile[2][DIMSZ * 16];           // 2 x 8 KB [d][col]
    const int wave = threadIdx.x >> 5;
    const int lane = threadIdx.x & 31;
    const int hi   = lane >> 4;                     // half-wave 0/1
    const int lm   = lane & 15;                     // M (A) / N (B,C)
    const int rowBase = blockIdx.x * 128 + wave * 16;
    const int row = rowBase + lm;

    // A fragments: full 16x128 row tile in registers (loaded once).
    v2f a[32];
    const float* xrow = x + (size_t)row * DIMSZ + 2 * hi;
    #pragma unroll
    for (int c = 0; c < 32; ++c) a[c] = *(const v2f*)(xrow + 4 * c);

    // Tile copy addressing: 512 float4 per tile, 2 per thread.
    const int d0  = threadIdx.x >> 2;
    const int c40 = (threadIdx.x & 3) << 2;
    const int d1  = d0 + 64;
    const float* g0 = embed + d0 * KCB + c40;
    const float* g1 = embed + d1 * KCB + c40;
    float* l0a = &tile[0][d0 * 16 + c40];
    float* l1a = &tile[0][d1 * 16 + c40];
    float* l0b = &tile[1][d0 * 16 + c40];
    float* l1b = &tile[1][d1 * 16 + c40];

    // Prologue: tile 0 -> buf 0
    v4f s0 = *(const v4f*)g0;
    v4f s1 = *(const v4f*)g1;
    *(v4f*)l0a = s0;
    *(v4f*)l1a = s1;
    __syncthreads();

    float best[8];
    int   bidx[8];
    #pragma unroll
    for (int r = 0; r < 8; ++r) { best[r] = -3.4e38f; bidx[r] = 0; }

    for (int t = 0; t < KCB / 16; ++t) {
        const int cur = t & 1;
        if (t < KCB / 16 - 1) {                     // issue prefetch of tile t+1
            s0 = *(const v4f*)(g0 + (t + 1) * 16);
            s1 = *(const v4f*)(g1 + (t + 1) * 16);
        }

        const float* lb = tile[cur];
        v2f bfA[8], bfB[8];
        #pragma unroll
        for (int c = 0; c < 8; ++c) {               // batch 0
            const int dbase = 4 * c + 2 * hi;
            bfA[c].x = lb[(dbase + 0) * 16 + lm];
            bfA[c].y = lb[(dbase + 1) * 16 + lm];
        }
        SCHED_FENCE();

        v8f acc0 = {0.f, 0.f, 0.f, 0.f, 0.f, 0.f, 0.f, 0.f};
        v8f acc1 = {0.f, 0.f, 0.f, 0.f, 0.f, 0.f, 0.f, 0.f};
        #pragma unroll
        for (int g = 0; g < 4; ++g) {
            v2f* curb = (g & 1) ? bfB : bfA;
            v2f* nxtb = (g & 1) ? bfA : bfB;
            if (g < 3) {                            // load batch g+1 ahead
                #pragma unroll
                for (int c = 0; c < 8; ++c) {
                    const int dbase = 4 * ((g + 1) * 8 + c) + 2 * hi;
                    nxtb[c].x = lb[(dbase + 0) * 16 + lm];
                    nxtb[c].y = lb[(dbase + 1) * 16 + lm];
                }
            }
            SCHED_FENCE();
            #pragma unroll
            for (int c = 0; c < 8; c += 2) {        // compute batch g
                acc0 = __builtin_amdgcn_wmma_f32_16x16x4_f32(
                    false, a[g * 8 + c + 0], false, curb[c + 0], (short)0, acc0, false, false);
                acc1 = __builtin_amdgcn_wmma_f32_16x16x4_f32(
                    false, a[g * 8 + c + 1], false, curb[c + 1], (short)0, acc1, false, false);
            }
            SCHED_FENCE();
        }

        const float hn  = half_norm[t * 16 + lm];
        const int   col = t * 16 + lm;
        #pragma unroll
        for (int r = 0; r < 8; ++r) {
            float sc = (acc0[r] + acc1[r]) - hn;    // maximize x.e - 0.5|e|^2
            if (sc > best[r]) { best[r] = sc; bidx[r] = col; }
        }

        if (t < KCB / 16 - 1) {                     // drain prefetch into next buf
            *(v4f*)(cur ? l0a : l0b) = s0;
            *(v4f*)(cur ? l1a : l1b) = s1;
        }
        __syncthreads();
    }

    // argmax across the 16 lanes of each half-wave (xor 1,2,4,8 stays in half)
    #pragma unroll
    for (int m = 1; m < 16; m <<= 1) {
        #pragma unroll
        for (int r = 0; r < 8; ++r) {
            float ov = __shfl_xor(best[r], m, 32);
            int   oi = __shfl_xor(bidx[r], m, 32);
            if (ov > best[r] || (ov == best[r] && oi < bidx[r])) { best[r] = ov; bidx[r] = oi; }
        }
    }
    if (lm == 0) {
        const int rBase = rowBase + hi * 8;         // lanes 0/16 own rows 0-7 / 8-15
        #pragma unroll
        for (int r = 0; r < 8; ++r) {
            ind_out[rBase + r] = (float)bidx[r];
            atomicAdd(&cnt[bidx[r]], 1.0f);
        }
    }
}

// Gather quantize, loss partial, scatter embed_sum.
__global__ __launch_bounds__(128)
void k_quant(const float* __restrict__ x, const float* __restrict__ embed,
             const float* __restrict__ ind_f, float* __restrict__ q_out,
             float* __restrict__ embed_sum, float* __restrict__ loss_acc) {
    const int d = threadIdx.x;
    const int rowBase = blockIdx.x * 8;
    float lsum = 0.0f;
    #pragma unroll
    for (int r = 0; r < 8; ++r) {
        const int n = rowBase + r;
        const int k = (int)ind_f[n];
        float q  = embed[d * KCB + k];              // 2MB table, L2-resident
        float xv = x[(size_t)n * DIMSZ + d];
        q_out[(size_t)n * DIMSZ + d] = q;           // quantize_st == quantize
        float df = q - xv;
        lsum += df * df;
        atomicAdd(&embed_sum[d * KCB + k], xv);
    }
    #pragma unroll
    for (int m = 16; m >= 1; m >>= 1) lsum += __shfl_xor(lsum, m, 32);
    __shared__ float red[4];
    if ((threadIdx.x & 31) == 0) red[threadIdx.x >> 5] = lsum;
    __syncthreads();
    if (threadIdx.x == 0) atomicAdd(loss_acc, red[0] + red[1] + red[2] + red[3]);
}

__global__ void k_cluster(const float* __restrict__ cs, const float* __restrict__ cnt,
                          float* __restrict__ cs_new_out,
                          float* __restrict__ n_acc, float* __restrict__ small_acc) {
    const int k = blockIdx.x * 256 + threadIdx.x;
    float v = cs[k] * 0.99f + 0.01f * cnt[k];
    cs_new_out[k] = v;
    float s1 = v, s2 = (v < 1.0f) ? 1.0f : 0.0f;
    #pragma unroll
    for (int m = 16; m >= 1; m >>= 1) { s1 += __shfl_xor(s1, m, 32); s2 += __shfl_xor(s2, m, 32); }
    __shared__ float r1[8], r2[8];
    const int w = threadIdx.x >> 5;
    if ((threadIdx.x & 31) == 0) { r1[w] = s1; r2[w] = s2; }
    __syncthreads();
    if (threadIdx.x == 0) {
        float t1 = 0.f, t2 = 0.f;
        for (int i = 0; i < 8; ++i) { t1 += r1[i]; t2 += r2[i]; }
        atomicAdd(n_acc, t1); atomicAdd(small_acc, t2);
    }
}

__global__ void k_final(const float* __restrict__ embed_avg, const float* __restrict__ embed_sum,
                        const float* __restrict__ cs_new, const float* __restrict__ acc,
                        float* __restrict__ embed_new_out, float* __restrict__ embed_avg_new_out,
                        float* __restrict__ loss_out, float* __restrict__ small_out) {
    const int i = blockIdx.x * 256 + threadIdx.x;
    float ea = embed_avg[i] * 0.99f + 0.01f * embed_sum[i];
    embed_avg_new_out[i] = ea;
    const int k = i & (KCB - 1);
    const float n = acc[1];
    const float csn = (cs_new[k] + 1e-5f) / (n + KCB * 1e-5f) * n;
    embed_new_out[i] = ea / csn;
    if (i == 0) {
        *loss_out  = acc[0] / (float)(NROWS * DIMSZ);
        *small_out = acc[2];
    }
}

extern "C" void kernel_launch(void* const* d_in, const int* in_sizes, int n_in,
                              void* d_out, int out_size, void* d_ws, size_t ws_size,
                              hipStream_t stream) {
    (void)in_sizes; (void)n_in; (void)out_size; (void)ws_size;
    const float* x            = (const float*)d_in[0];
    const float* embed        = (const float*)d_in[1];
    const float* cluster_size = (const float*)d_in[2];
    const float* embed_avg    = (const float*)d_in[3];

    float* out = (float*)d_out;
    float* ws  = (float*)d_ws;
    float* half_norm = ws;
    float* cnt       = ws + WS_CNT;
    float* esum      = ws + WS_ESUM;
    float* acc       = ws + WS_ACC;

    k_init   <<<(WS_ZERO_LEN + 255) / 256, 256, 0, stream>>>(ws);
    k_norm   <<<KCB / 256, 256, 0, stream>>>(embed, half_norm);
    k_assign <<<NROWS / 128, 256, 0, stream>>>(x, embed, half_norm, out + O_IND, cnt);
    k_quant  <<<NROWS / 8, 128, 0, stream>>>(x, embed, out + O_IND, out + O_Q, esum, acc);
    k_cluster<<<KCB / 256, 256, 0, stream>>>(cluster_size, cnt, out + O_CSNEW, acc + 1, acc + 2);
    k_final  <<<(DIMSZ * KCB) / 256, 256, 0, stream>>>(embed_avg, esum, out + O_CSNEW, acc,
                                                       out + O_ENEW, out + O_EANEW,
                                                       out + O_LOSS, out + O_SMALL);
}